// MoE_1821066134126
// MI455X (gfx1250) — compile-verified
//
#include <hip/hip_runtime.h>
#include <hip/hip_bf16.h>
#include <cstdint>

// ---------------------------------------------------------------------------
// MI455X (gfx1250) implementation.
//  S=1024, B=4, D=1024, H=16, DH=64, E=3, T=4096.
//  Live outputs: out = (x + attn(x)) * gate_score   and   router_logits.
//  The expert FFN in the reference is dead code (result deleted) -> skipped.
//  All matmuls use V_WMMA_F32_16X16X4_F32 (fp32-exact).  Tile staging uses
//  GLOBAL_LOAD_ASYNC_TO_LDS_B128 (ASYNCcnt-tracked, no VGPR round-trip).
// ---------------------------------------------------------------------------

typedef __attribute__((ext_vector_type(2))) float v2f;
typedef __attribute__((ext_vector_type(8))) float v8f;
typedef __attribute__((ext_vector_type(4))) int   v4i;

#define SEQ   1024
#define BAT   4
#define DIM   1024
#define NHEAD 16
#define DHEAD 64
#define NTOK  4096   // S*B

#define AS1 __attribute__((address_space(1)))
#define AS3 __attribute__((address_space(3)))

#if defined(__has_builtin)
#if __has_builtin(__builtin_amdgcn_global_load_async_to_lds_b128)
#define HAVE_ASYNC_LDS 1
#endif
#endif

// Copy 16B from global to LDS.  Async (ASYNCcnt) when available.
static __device__ __forceinline__ void copy_b128_to_lds(const float* g,
                                                        float* l) {
#ifdef HAVE_ASYNC_LDS
  __builtin_amdgcn_global_load_async_to_lds_b128(
      (AS1 v4i*)(uintptr_t)g,
      (AS3 v4i*)(uint32_t)(uintptr_t)l, 0, 0);
#else
  *(float4*)l = *(const float4*)g;
#endif
}

static __device__ __forceinline__ void async_wait0() {
#ifdef HAVE_ASYNC_LDS
#if __has_builtin(__builtin_amdgcn_s_wait_asynccnt)
  __builtin_amdgcn_s_wait_asynccnt(0);
#else
  asm volatile("s_wait_asynccnt 0x0" ::: "memory");
#endif
#endif
}

static __device__ __forceinline__ float wave_sum(float v) {
#pragma unroll
  for (int m = 1; m < 32; m <<= 1) v += __shfl_xor(v, m, 32);
  return v;
}

// ---------------------------------------------------------------------------
// Kernel 1: LayerNorm (per token).
// ---------------------------------------------------------------------------
__global__ __launch_bounds__(256) void ln_kernel(
    const float* __restrict__ x, const float* __restrict__ w,
    const float* __restrict__ b, float* __restrict__ out) {
  const int t = blockIdx.x, tid = threadIdx.x;
  const float* row = x + (size_t)t * DIM;
  float4 v = ((const float4*)row)[tid];
  float s  = v.x + v.y + v.z + v.w;
  float s2 = v.x * v.x + v.y * v.y + v.z * v.z + v.w * v.w;
  s = wave_sum(s); s2 = wave_sum(s2);
  __shared__ float red[8][2];
  const int wv = tid >> 5, ln = tid & 31;
  if (ln == 0) { red[wv][0] = s; red[wv][1] = s2; }
  __syncthreads();
  float tot = 0.f, tot2 = 0.f;
#pragma unroll
  for (int i = 0; i < 8; ++i) { tot += red[i][0]; tot2 += red[i][1]; }
  const float mu = tot * (1.f / DIM);
  const float rstd = rsqrtf(tot2 * (1.f / DIM) - mu * mu + 1e-5f);
  float4 w4 = ((const float4*)w)[tid];
  float4 b4 = ((const float4*)b)[tid];
  float4 o;
  o.x = (v.x - mu) * rstd * w4.x + b4.x;
  o.y = (v.y - mu) * rstd * w4.y + b4.y;
  o.z = (v.z - mu) * rstd * w4.z + b4.z;
  o.w = (v.w - mu) * rstd * w4.w + b4.w;
  ((float4*)(out + (size_t)t * DIM))[tid] = o;
}

// ---------------------------------------------------------------------------
// WMMA GEMM tiling: block = 4 waves (128 thr), block tile 128(M)x64(N),
// K-chunk 32 staged in LDS; each wave computes 32x64 = 2x4 fragments.
// ds/wmma ratio: 6 ds_load_b64 per 8 v_wmma.
// ---------------------------------------------------------------------------
#define BM 128
#define BN 64
#define KC 32
#define LDSW 40   // padded row stride (160B: 16B-aligned rows, 8B frags)

// Kernel 2: QKV GEMM. C[T,3D] = h @ in_proj_w^T + b, scattered to q/k/v
// buffers in [B,H,S,DH] layout (q scaled by DH^-0.5).
__global__ __launch_bounds__(128) void qkv_gemm_kernel(
    const float* __restrict__ A, const float* __restrict__ W,
    const float* __restrict__ bias, float* __restrict__ qp,
    float* __restrict__ kp, float* __restrict__ vp) {
  __shared__ float As[BM * LDSW];
  __shared__ float Bs[BN * LDSW];
  const int tid = threadIdx.x;
  const int n0 = blockIdx.x * BN;
  const int m0 = blockIdx.y * BM;
  const int wv = tid >> 5, lane = tid & 31;
  const int hl = lane >> 4, lc = lane & 15;
  const int mw = wv * 32;

  v8f acc[2][4];
#pragma unroll
  for (int f = 0; f < 4; ++f) {
    const float bv = bias[n0 + f * 16 + lc];
#pragma unroll
    for (int i = 0; i < 8; ++i) { acc[0][f][i] = bv; acc[1][f][i] = bv; }
  }

  const int srow = tid >> 3;        // 0..15
  const int sseg = (tid & 7) * 4;   // 0..28 step 4

  for (int k0 = 0; k0 < DIM; k0 += KC) {
#pragma unroll
    for (int j = 0; j < 8; ++j) {   // A: 128 rows
      const int r = j * 16 + srow;
      copy_b128_to_lds(&A[(size_t)(m0 + r) * DIM + k0 + sseg],
                       &As[r * LDSW + sseg]);
    }
#pragma unroll
    for (int j = 0; j < 4; ++j) {   // B: 64 rows
      const int r = j * 16 + srow;
      copy_b128_to_lds(&W[(size_t)(n0 + r) * DIM + k0 + sseg],
                       &Bs[r * LDSW + sseg]);
    }
    async_wait0();
    __syncthreads();
#pragma unroll
    for (int kk = 0; kk < KC; kk += 4) {
      v2f a0 = *(const v2f*)&As[(mw + lc) * LDSW + kk + 2 * hl];
      v2f a1 = *(const v2f*)&As[(mw + 16 + lc) * LDSW + kk + 2 * hl];
#pragma unroll
      for (int f = 0; f < 4; ++f) {
        v2f b = *(const v2f*)&Bs[(f * 16 + lc) * LDSW + kk + 2 * hl];
        acc[0][f] = __builtin_amdgcn_wmma_f32_16x16x4_f32(
            false, a0, false, b, (short)0, acc[0][f], false, false);
        acc[1][f] = __builtin_amdgcn_wmma_f32_16x16x4_f32(
            false, a1, false, b, (short)0, acc[1][f], false, false);
      }
    }
    __syncthreads();
  }

  // n0 is 64-aligned: one (which, head) per block column.
  const int which = n0 >> 10;          // 0:q 1:k 2:v
  const int head  = (n0 >> 6) & 15;
  const float scale = (which == 0) ? 0.125f : 1.0f;   // DH^-0.5
  float* dst = (which == 0) ? qp : (which == 1 ? kp : vp);
#pragma unroll
  for (int mi = 0; mi < 2; ++mi) {
#pragma unroll
    for (int f = 0; f < 4; ++f) {
      const int dh = f * 16 + lc;
#pragma unroll
      for (int i = 0; i < 8; ++i) {
        const int m = m0 + mw + mi * 16 + i + 8 * hl;   // token index
        const int s_idx = m >> 2, bbi = m & 3;          // t = s*B + b
        dst[(((size_t)(bbi * NHEAD + head)) * SEQ + s_idx) * DHEAD + dh] =
            acc[mi][f][i] * scale;
      }
    }
  }
}

// Kernel 4: out_proj GEMM with fused residual: x2 = x + o @ W^T + b
__global__ __launch_bounds__(128) void outproj_gemm_kernel(
    const float* __restrict__ A, const float* __restrict__ W,
    const float* __restrict__ bias, const float* __restrict__ xres,
    float* __restrict__ x2) {
  __shared__ float As[BM * LDSW];
  __shared__ float Bs[BN * LDSW];
  const int tid = threadIdx.x;
  const int n0 = blockIdx.x * BN;
  const int m0 = blockIdx.y * BM;
  const int wv = tid >> 5, lane = tid & 31;
  const int hl = lane >> 4, lc = lane & 15;
  const int mw = wv * 32;

  v8f acc[2][4];
#pragma unroll
  for (int f = 0; f < 4; ++f) {
    const int n_glob = n0 + f * 16 + lc;
    const float bv = bias[n_glob];
#pragma unroll
    for (int mi = 0; mi < 2; ++mi)
#pragma unroll
      for (int i = 0; i < 8; ++i) {
        const int m = m0 + mw + mi * 16 + i + 8 * hl;
        acc[mi][f][i] = bv + xres[(size_t)m * DIM + n_glob];  // residual
      }
  }

  const int srow = tid >> 3;
  const int sseg = (tid & 7) * 4;

  for (int k0 = 0; k0 < DIM; k0 += KC) {
#pragma unroll
    for (int j = 0; j < 8; ++j) {
      const int r = j * 16 + srow;
      copy_b128_to_lds(&A[(size_t)(m0 + r) * DIM + k0 + sseg],
                       &As[r * LDSW + sseg]);
    }
#pragma unroll
    for (int j = 0; j < 4; ++j) {
      const int r = j * 16 + srow;
      copy_b128_to_lds(&W[(size_t)(n0 + r) * DIM + k0 + sseg],
                       &Bs[r * LDSW + sseg]);
    }
    async_wait0();
    __syncthreads();
#pragma unroll
    for (int kk = 0; kk < KC; kk += 4) {
      v2f a0 = *(const v2f*)&As[(mw + lc) * LDSW + kk + 2 * hl];
      v2f a1 = *(const v2f*)&As[(mw + 16 + lc) * LDSW + kk + 2 * hl];
#pragma unroll
      for (int f = 0; f < 4; ++f) {
        v2f b = *(const v2f*)&Bs[(f * 16 + lc) * LDSW + kk + 2 * hl];
        acc[0][f] = __builtin_amdgcn_wmma_f32_16x16x4_f32(
            false, a0, false, b, (short)0, acc[0][f], false, false);
        acc[1][f] = __builtin_amdgcn_wmma_f32_16x16x4_f32(
            false, a1, false, b, (short)0, acc[1][f], false, false);
      }
    }
    __syncthreads();
  }

#pragma unroll
  for (int mi = 0; mi < 2; ++mi)
#pragma unroll
    for (int f = 0; f < 4; ++f)
#pragma unroll
      for (int i = 0; i < 8; ++i) {
        const int m = m0 + mw + mi * 16 + i + 8 * hl;
        x2[(size_t)m * DIM + n0 + f * 16 + lc] = acc[mi][f][i];
      }
}

// ---------------------------------------------------------------------------
// Kernel 3: causal flash attention.  One wave per (b, h, 16-row query block).
// Q, K, V tiles staged in LDS via the async path; Q@K^T and P@V via fp32
// WMMA; online softmax in the C-fragment layout (16-lane shfl reductions);
// P re-laid-out through LDS into an A fragment.
// ---------------------------------------------------------------------------
__global__ __launch_bounds__(32) void attn_kernel(
    const float* __restrict__ q, const float* __restrict__ k,
    const float* __restrict__ v, float* __restrict__ o /*[S,B,D]*/) {
  const int lane = threadIdx.x;
  const int bid = blockIdx.x;
  const int qb = bid & 63;
  const int hh = (bid >> 6) & 15;
  const int bbi = bid >> 10;
  const int q0 = qb << 4;
  const size_t hb = ((size_t)(bbi * NHEAD + hh)) * SEQ * DHEAD;
  const float* qh = q + hb;
  const float* kh = k + hb;
  const float* vh = v + hb;

  __shared__ float Qs[16 * 64];
  __shared__ float Ks[16 * 64];
  __shared__ float Vs[16 * 64];
  __shared__ float Ps[16 * 16];

#pragma unroll
  for (int it = 0; it < 8; ++it) {   // stage 16x64 Q tile
    const int flat = it * 128 + lane * 4;
    copy_b128_to_lds(&qh[(size_t)(q0 + (flat >> 6)) * DHEAD + (flat & 63)],
                     &Qs[flat]);
  }

  const int hl = lane >> 4, lc = lane & 15;
  const float NEG_INF = -__builtin_huge_valf();
  float mrow[8], lrow[8];
#pragma unroll
  for (int i = 0; i < 8; ++i) { mrow[i] = NEG_INF; lrow[i] = 0.f; }
  v8f acc[4];
#pragma unroll
  for (int f = 0; f < 4; ++f)
#pragma unroll
    for (int i = 0; i < 8; ++i) acc[f][i] = 0.f;

  for (int jb = 0; jb <= qb; ++jb) {
    const int j0 = jb << 4;
#pragma unroll
    for (int it = 0; it < 8; ++it) {   // stage 16x64 K and V tiles
      const int flat = it * 128 + lane * 4;
      const size_t goff = (size_t)(j0 + (flat >> 6)) * DHEAD + (flat & 63);
      copy_b128_to_lds(&kh[goff], &Ks[flat]);
      copy_b128_to_lds(&vh[goff], &Vs[flat]);
    }
    async_wait0();
    __syncthreads();

    v8f sc;
#pragma unroll
    for (int i = 0; i < 8; ++i) sc[i] = 0.f;
#pragma unroll
    for (int k0 = 0; k0 < DHEAD; k0 += 4) {
      v2f a = *(const v2f*)&Qs[lc * 64 + k0 + 2 * hl];
      v2f b = *(const v2f*)&Ks[lc * 64 + k0 + 2 * hl];
      sc = __builtin_amdgcn_wmma_f32_16x16x4_f32(false, a, false, b,
                                                 (short)0, sc, false, false);
    }

    const bool diag = (jb == qb);
    float p[8], alpha[8];
#pragma unroll
    for (int i = 0; i < 8; ++i) {
      const int r = i + 8 * hl;
      float s = sc[i];
      if (diag && (lc > r)) s = NEG_INF;   // causal mask on diagonal block
      float rm = s;
      rm = fmaxf(rm, __shfl_xor(rm, 1, 32));
      rm = fmaxf(rm, __shfl_xor(rm, 2, 32));
      rm = fmaxf(rm, __shfl_xor(rm, 4, 32));
      rm = fmaxf(rm, __shfl_xor(rm, 8, 32));
      const float mnew = fmaxf(mrow[i], rm);
      alpha[i] = __expf(mrow[i] - mnew);
      mrow[i] = mnew;
      const float pe = __expf(s - mnew);
      float rs = pe;
      rs += __shfl_xor(rs, 1, 32);
      rs += __shfl_xor(rs, 2, 32);
      rs += __shfl_xor(rs, 4, 32);
      rs += __shfl_xor(rs, 8, 32);
      lrow[i] = lrow[i] * alpha[i] + rs;
      p[i] = pe;
    }
#pragma unroll
    for (int f = 0; f < 4; ++f)
#pragma unroll
      for (int i = 0; i < 8; ++i) acc[f][i] *= alpha[i];

#pragma unroll
    for (int i = 0; i < 8; ++i) Ps[(i + 8 * hl) * 16 + lc] = p[i];
    __syncthreads();   // single wave: LDS ops are in-order; barrier is cheap

#pragma unroll
    for (int f = 0; f < 4; ++f) {
      const int n0 = f * 16;
#pragma unroll
      for (int k0 = 0; k0 < 16; k0 += 4) {
        v2f a = *(const v2f*)&Ps[lc * 16 + k0 + 2 * hl];
        const int kk = k0 + 2 * hl;
        v2f b;
        b.x = Vs[kk * 64 + n0 + lc];
        b.y = Vs[(kk + 1) * 64 + n0 + lc];
        acc[f] = __builtin_amdgcn_wmma_f32_16x16x4_f32(
            false, a, false, b, (short)0, acc[f], false, false);
      }
    }
    __syncthreads();   // Ks/Vs/Ps reused next iteration
  }

#pragma unroll
  for (int f = 0; f < 4; ++f) {
#pragma unroll
    for (int i = 0; i < 8; ++i) {
      const int r = i + 8 * hl;
      const int s_idx = q0 + r;
      const int col = hh * DHEAD + f * 16 + lc;
      o[((size_t)s_idx * BAT + bbi) * DIM + col] = acc[f][i] / lrow[i];
    }
  }
}

// ---------------------------------------------------------------------------
// Kernel 5: LN2 + router logits + top-2 gate + final scaled output.
//   gate = r_max / (1 - r_min)  where r = softmax(logits), E = 3.
// ---------------------------------------------------------------------------
__global__ __launch_bounds__(256) void router_out_kernel(
    const float* __restrict__ x2, const float* __restrict__ w,
    const float* __restrict__ b, const float* __restrict__ gw,
    float* __restrict__ out, float* __restrict__ logits) {
  const int t = blockIdx.x, tid = threadIdx.x;
  const float* row = x2 + (size_t)t * DIM;
  float4 v = ((const float4*)row)[tid];
  float s  = v.x + v.y + v.z + v.w;
  float s2 = v.x * v.x + v.y * v.y + v.z * v.z + v.w * v.w;
  s = wave_sum(s); s2 = wave_sum(s2);
  __shared__ float red[8][5];
  const int wv = tid >> 5, ln = tid & 31;
  if (ln == 0) { red[wv][0] = s; red[wv][1] = s2; }
  __syncthreads();
  float tot = 0.f, tot2 = 0.f;
#pragma unroll
  for (int i = 0; i < 8; ++i) { tot += red[i][0]; tot2 += red[i][1]; }
  const float mu = tot * (1.f / DIM);
  const float rstd = rsqrtf(tot2 * (1.f / DIM) - mu * mu + 1e-5f);
  float4 w4 = ((const float4*)w)[tid];
  float4 b4 = ((const float4*)b)[tid];
  float4 hn;
  hn.x = (v.x - mu) * rstd * w4.x + b4.x;
  hn.y = (v.y - mu) * rstd * w4.y + b4.y;
  hn.z = (v.z - mu) * rstd * w4.z + b4.z;
  hn.w = (v.w - mu) * rstd * w4.w + b4.w;
  float part[3];
#pragma unroll
  for (int e = 0; e < 3; ++e) {
    float4 g = ((const float4*)(gw + (size_t)e * DIM))[tid];
    part[e] = hn.x * g.x + hn.y * g.y + hn.z * g.z + hn.w * g.w;
    part[e] = wave_sum(part[e]);
  }
  if (ln == 0) { red[wv][2] = part[0]; red[wv][3] = part[1]; red[wv][4] = part[2]; }
  __syncthreads();
  float l0 = 0.f, l1 = 0.f, l2 = 0.f;
#pragma unroll
  for (int i = 0; i < 8; ++i) { l0 += red[i][2]; l1 += red[i][3]; l2 += red[i][4]; }
  if (tid == 0) {
    logits[t * 3 + 0] = l0; logits[t * 3 + 1] = l1; logits[t * 3 + 2] = l2;
  }
  const float mx = fmaxf(l0, fmaxf(l1, l2));
  const float e0 = __expf(l0 - mx), e1 = __expf(l1 - mx), e2 = __expf(l2 - mx);
  const float inv = 1.f / (e0 + e1 + e2);
  const float r0 = e0 * inv, r1 = e1 * inv, r2 = e2 * inv;
  const float rmax = fmaxf(r0, fmaxf(r1, r2));
  const float rmin = fminf(r0, fminf(r1, r2));
  const float gate = rmax / (1.f - rmin);   // top-2 renormalized max
  float4 ov;
  ov.x = v.x * gate; ov.y = v.y * gate; ov.z = v.z * gate; ov.w = v.w * gate;
  ((float4*)(out + (size_t)t * DIM))[tid] = ov;
}

// ---------------------------------------------------------------------------
extern "C" void kernel_launch(void* const* d_in, const int* in_sizes, int n_in,
                              void* d_out, int out_size, void* d_ws,
                              size_t ws_size, hipStream_t stream) {
  (void)in_sizes; (void)n_in; (void)out_size; (void)ws_size;
  const float* x         = (const float*)d_in[0];
  const float* ln1_w     = (const float*)d_in[1];
  const float* ln1_b     = (const float*)d_in[2];
  const float* ln2_w     = (const float*)d_in[3];
  const float* ln2_b     = (const float*)d_in[4];
  const float* in_proj_w = (const float*)d_in[5];
  const float* in_proj_b = (const float*)d_in[6];
  const float* out_proj_w= (const float*)d_in[7];
  const float* out_proj_b= (const float*)d_in[8];
  const float* gate_w    = (const float*)d_in[9];
  // d_in[10..13] (fc_w, fc_b, proj_w, proj_b): dead code in the reference.

  const size_t TD = (size_t)NTOK * DIM;   // 4,194,304 floats = 16 MB
  float* ws = (float*)d_ws;
  float* h_buf = ws;            // LN1 out; reused as attention output `o`
  float* q_buf = ws + TD;       // reused as x2 after attention consumes q
  float* k_buf = ws + 2 * TD;
  float* v_buf = ws + 3 * TD;   // total workspace: 64 MB
  float* o_buf = h_buf;
  float* x2    = q_buf;

  float* out_f  = (float*)d_out;
  float* logits = out_f + TD;

  ln_kernel<<<NTOK, 256, 0, stream>>>(x, ln1_w, ln1_b, h_buf);
  qkv_gemm_kernel<<<dim3(48, 32), 128, 0, stream>>>(h_buf, in_proj_w,
                                                    in_proj_b, q_buf, k_buf,
                                                    v_buf);
  attn_kernel<<<NTOK, 32, 0, stream>>>(q_buf, k_buf, v_buf, o_buf);
  outproj_gemm_kernel<<<dim3(16, 32), 128, 0, stream>>>(o_buf, out_proj_w,
                                                        out_proj_b, x, x2);
  router_out_kernel<<<NTOK, 256, 0, stream>>>(x2, ln2_w, ln2_b, gate_w, out_f,
                                              logits);
}